// StridedSparseAttention_90640989815088
// MI455X (gfx1250) — compile-verified
//
#include <hip/hip_runtime.h>
#include <math.h>

typedef __attribute__((ext_vector_type(2))) float v2f;
typedef __attribute__((ext_vector_type(4))) float v4f;
typedef __attribute__((ext_vector_type(8))) float v8f;

#define TT      2048
#define DH      64
#define NSTR    128          // TT/16 stride keys
#define KS_STR  68           // LDS row stride for K/V tiles (16B aligned, bank-skewed)
#define P_STR   148          // LDS row stride for P (>=144, even, 20-bank skew)

__global__ __launch_bounds__(256)
void sattn_kernel(const float* __restrict__ q, const float* __restrict__ k,
                  const float* __restrict__ v, const int* __restrict__ mask,
                  float* __restrict__ out, float* __restrict__ attn)
{
    extern __shared__ __align__(16) float smem[];
    float* Ks   = smem;                        // [128][68]
    float* Vs   = Ks + NSTR * KS_STR;          // [128][68]
    float* Pall = Vs + NSTR * KS_STR;          // [8][16][148]
    int*   Msk  = (int*)(Pall + 8 * 16 * P_STR); // [128]

    const int tid  = threadIdx.x;
    const int wave = tid >> 5;
    const int lane = tid & 31;
    const int n    = lane & 15;
    const int half = lane >> 4;

    const int bh  = blockIdx.x >> 4;           // 0..31  (b*16+h)
    const int qb  = blockIdx.x & 15;           // 0..15
    const int b   = bh >> 4;
    const int q0w = qb * 128 + wave * 16;

    const size_t bhbase = (size_t)bh * TT * DH;
    const float* qp = q + bhbase;
    const float* kp = k + bhbase;
    const float* vp = v + bhbase;

    // ---- cooperative load: 128 stride K/V rows (rk = 16j) into LDS ----
    for (int idx = tid; idx < NSTR * (DH/4); idx += 256) {
        int j  = idx >> 4;
        int f4 = idx & 15;
        v4f kv = *(const v4f*)(kp + (size_t)(16*j) * DH + 4*f4);
        v4f vv = *(const v4f*)(vp + (size_t)(16*j) * DH + 4*f4);
        *(v4f*)(Ks + j*KS_STR + 4*f4) = kv;
        *(v4f*)(Vs + j*KS_STR + 4*f4) = vv;
    }
    if (tid < NSTR) Msk[tid] = mask[(size_t)b * TT + 16*tid];
    __syncthreads();

    // ==== Stage 1: S(16x144) = Q(16x64) @ Ksel^T  via v_wmma_f32_16x16x4_f32 ====
    // A layout: row = lane%16, k = 4s + 2*half + vgpr.  B: col = lane%16, same k.
    v2f qa[16];
    #pragma unroll
    for (int s = 0; s < 16; ++s)
        qa[s] = *(const v2f*)(qp + (size_t)(q0w + n) * DH + 4*s + 2*half);

    v8f accS[9] = {};
    #pragma unroll
    for (int s = 0; s < 16; ++s) {
        v2f a = qa[s];
        #pragma unroll
        for (int c = 0; c < 8; ++c) {           // 8 stride column tiles
            v2f bv = *(const v2f*)(Ks + (16*c + n) * KS_STR + 4*s + 2*half);
            accS[c] = __builtin_amdgcn_wmma_f32_16x16x4_f32(
                false, a, false, bv, (short)0, accS[c], false, false);
        }
        { // diagonal tile: keys q0w..q0w+15 straight from global
            v2f bv = *(const v2f*)(kp + (size_t)(q0w + n) * DH + 4*s + 2*half);
            accS[8] = __builtin_amdgcn_wmma_f32_16x16x4_f32(
                false, a, false, bv, (short)0, accS[8], false, false);
        }
    }

    // ==== Stage 2: masked softmax over 128 stride cols + diagonal ====
    const float scale = 0.125f;                 // 1/sqrt(64)
    float* Pw = Pall + wave * 16 * P_STR;

    float mokf[8];
    #pragma unroll
    for (int c = 0; c < 8; ++c) mokf[c] = (Msk[16*c + n] != 0) ? 1.f : 0.f;

    #pragma unroll
    for (int vv = 0; vv < 8; ++vv) {
        const int row = vv + 8*half;            // query row within tile
        const int dm  = mask[(size_t)b * TT + q0w + row];
        // diag element valid only on lane n==row; row==0 duplicates stride col
        const bool dvalid = (n == row) && (row != 0) && (dm != 0);

        float sv[8], rowmax = -1e30f;
        #pragma unroll
        for (int c = 0; c < 8; ++c) {
            float s = accS[c][vv] * scale;
            sv[c] = (mokf[c] != 0.f) ? s : -1e30f;
            rowmax = fmaxf(rowmax, sv[c]);
        }
        const float dsc = accS[8][vv] * scale;
        rowmax = fmaxf(rowmax, dvalid ? dsc : -1e30f);
        #pragma unroll
        for (int off = 1; off < 16; off <<= 1)
            rowmax = fmaxf(rowmax, __shfl_xor(rowmax, off, 16));

        float sum = 0.f, pv[8];
        #pragma unroll
        for (int c = 0; c < 8; ++c) {
            pv[c] = (sv[c] > -1e29f) ? __expf(sv[c] - rowmax) : 0.f;
            sum += pv[c];
        }
        float pd = dvalid ? __expf(dsc - rowmax) : 0.f;
        sum += pd;
        #pragma unroll
        for (int off = 1; off < 16; off <<= 1)
            sum += __shfl_xor(sum, off, 16);
        const float inv = (sum > 0.f) ? (1.0f / sum) : 0.f;

        #pragma unroll
        for (int c = 0; c < 8; ++c) Pw[row * P_STR + 16*c + n] = pv[c] * inv;
        Pw[row * P_STR + 128 + n] = pd * inv;   // diag region (zeros off-diagonal)
    }

    // ==== Stage 3: out(16x64) = P(16x144) @ Vcat(144x64) via WMMA ====
    v8f accO[4] = {};
    for (int s = 0; s < 32; ++s) {              // stride-V rows from LDS
        v2f a = *(const v2f*)(Pw + n * P_STR + 4*s + 2*half);
        const int kk = 4*s + 2*half;
        #pragma unroll
        for (int ct = 0; ct < 4; ++ct) {
            v2f bv;
            bv.x = Vs[kk       * KS_STR + 16*ct + n];
            bv.y = Vs[(kk + 1) * KS_STR + 16*ct + n];
            accO[ct] = __builtin_amdgcn_wmma_f32_16x16x4_f32(
                false, a, false, bv, (short)0, accO[ct], false, false);
        }
    }
    #pragma unroll
    for (int s = 32; s < 36; ++s) {             // diagonal V rows from global
        v2f a = *(const v2f*)(Pw + n * P_STR + 4*s + 2*half);
        const int dk = 4*(s - 32) + 2*half;
        #pragma unroll
        for (int ct = 0; ct < 4; ++ct) {
            v2f bv;
            bv.x = vp[(size_t)(q0w + dk    ) * DH + 16*ct + n];
            bv.y = vp[(size_t)(q0w + dk + 1) * DH + 16*ct + n];
            accO[ct] = __builtin_amdgcn_wmma_f32_16x16x4_f32(
                false, a, false, bv, (short)0, accO[ct], false, false);
        }
    }
    #pragma unroll
    for (int vv = 0; vv < 8; ++vv) {
        const int row = vv + 8*half;
        float* op = out + bhbase + (size_t)(q0w + row) * DH;
        #pragma unroll
        for (int ct = 0; ct < 4; ++ct) op[16*ct + n] = accO[ct][vv];
    }

    // ==== Stage 4: stream sparse attn rows (16 x 2048) with NT float4 stores ====
    float* abase = attn + (size_t)bh * TT * TT;
    #pragma unroll 4
    for (int i = 0; i < 256; ++i) {
        const int idx = lane + 32*i;            // 0..8191 float4 slots
        const int row = idx >> 9;               // 16 rows * 512 float4
        const int q4  = idx & 511;
        const int col = q4 << 2;
        const int dpos = q0w + row;             // diagonal column

        v4f val = {0.f, 0.f, 0.f, 0.f};
        const bool hit = (q4 == (dpos >> 2));
        const float dvv = hit ? Pw[row * P_STR + 128 + row] : 0.f;
        const int dc = dpos & 3;
        val.x = (hit && dc == 0) ? dvv : 0.f;
        val.y = (hit && dc == 1) ? dvv : 0.f;
        val.z = (hit && dc == 2) ? dvv : 0.f;
        val.w = (hit && dc == 3) ? dvv : 0.f;
        if ((col & 15) == 0) val.x = Pw[row * P_STR + (col >> 4)]; // stride col (also fixes row==0 dup)

        __builtin_nontemporal_store(val, (v4f*)(abase + (size_t)dpos * TT + col));
    }
}

extern "C" void kernel_launch(void* const* d_in, const int* in_sizes, int n_in,
                              void* d_out, int out_size, void* d_ws, size_t ws_size,
                              hipStream_t stream) {
    (void)in_sizes; (void)n_in; (void)d_ws; (void)ws_size; (void)out_size;
    const float* q    = (const float*)d_in[0];
    const float* k    = (const float*)d_in[1];
    const float* v    = (const float*)d_in[2];
    const int*   mask = (const int*)d_in[3];
    float* out  = (float*)d_out;
    float* attn = out + (size_t)2 * 16 * 2048 * 64;   // out first, then attn

    const size_t smem = ((size_t)2 * NSTR * KS_STR + 8 * 16 * P_STR) * sizeof(float)
                        + NSTR * sizeof(int);         // ~146 KB (< 320 KB/WGP)
    dim3 grid(512), block(256);
    sattn_kernel<<<grid, block, smem, stream>>>(q, k, v, mask, out, attn);
}